// relpos_56358560858587
// MI455X (gfx1250) — compile-verified
//
#include <hip/hip_runtime.h>
#include <stdint.h>

// Problem constants from the reference
#define BB    2
#define NN    1024
#define NBINS 65
#define DIM   128

#define TAB_BYTES  (NBINS * DIM * 4)   // 33,280 B of W staged in LDS
#define TAB_CHUNKS (TAB_BYTES / 16)    // 2,080 x b128 chunks

typedef float v4f __attribute__((ext_vector_type(4)));

__global__ __launch_bounds__(256) void relpos_kernel(
    const int*   __restrict__ res_id,   // [B,N]
    const float* __restrict__ W,        // [65,128]
    const float* __restrict__ bia,      // [128]
    float*       __restrict__ out)      // [B,N,N,128]
{
    __shared__ float tab[NBINS * DIM];  // raw W table (bias lives in regs)

    const int tid  = threadIdx.x;
    const int lane = tid & 31;
    const int wave = tid >> 5;

    // ---- Stage W into LDS via CDNA5 async global->LDS copy (ASYNCcnt) ----
    {
        const uint32_t lds_base = (uint32_t)(uintptr_t)(&tab[0]); // low 32b of shared-aperture addr = LDS offset
        const uint64_t gbase    = (uint64_t)(uintptr_t)W;
        for (int c = tid; c < TAB_CHUNKS; c += 256) {
            uint32_t laddr = lds_base + (uint32_t)c * 16u;
            uint64_t gaddr = gbase + (uint64_t)c * 16u;
            asm volatile("global_load_async_to_lds_b128 %0, %1, off"
                         :: "v"(laddr), "v"(gaddr) : "memory");
        }
        asm volatile("s_wait_asynccnt 0x0" ::: "memory");
    }

    // Per-lane bias: lane owns dims [lane*4, lane*4+4) for every j (all 8 waves identical)
    v4f bias = *(const v4f*)(bia + lane * 4);

    __syncthreads();  // each wave waited on its own ASYNCcnt; barrier makes all LDS writes visible

    const int r  = blockIdx.x;        // flat (b,i) row, 0..2047
    const int bi = r >> 10;           // / N
    const int i  = r & (NN - 1);      // % N
    const int ri = res_id[bi * NN + i];

    float* __restrict__ orow = out + (size_t)r * (size_t)(NN * DIM);

    // Each wave handles one j per step: 32 lanes x float4 = 512 B contiguous NT store.
    #pragma unroll 4
    for (int j = wave; j < NN; j += 8) {
        int rj = res_id[bi * NN + j];            // wave-uniform broadcast load
        int d  = ri - rj;
        d = d < -32 ? -32 : (d > 32 ? 32 : d);   // clamp
        d += 32;                                  // 0..64

        // Bank-conflict-free ds_load_b128: d is wave-uniform, lane*16B spans all 64 banks
        v4f v = *(const v4f*)(tab + d * DIM + lane * 4);
        v += bias;

        // 1 GiB output >> 192 MB L2: stream past cache
        __builtin_nontemporal_store(v, (v4f*)(orow + (size_t)j * DIM + (size_t)lane * 4));
    }
}

extern "C" void kernel_launch(void* const* d_in, const int* in_sizes, int n_in,
                              void* d_out, int out_size, void* d_ws, size_t ws_size,
                              hipStream_t stream) {
    const int*   res_id = (const int*)d_in[0];   // [B,N] indices
    const float* W      = (const float*)d_in[1]; // [65,128]
    const float* bia    = (const float*)d_in[2]; // [128]
    float*       out    = (float*)d_out;         // [B,N,N,128] f32

    dim3 grid(BB * NN);   // 2048 blocks, one per (b,i) row (512 KB of output each)
    dim3 block(256);      // 8 waves (wave32)
    hipLaunchKernelGGL(relpos_kernel, grid, block, 0, stream, res_id, W, bia, out);
}